// QFF1_12223476924829
// MI455X (gfx1250) — compile-verified
//
#include <hip/hip_runtime.h>
#include <hip/hip_fp16.h>

// ---------------------------------------------------------------------------
// QFF1 encoder for MI455X (gfx1250, wave32).
//   points (N,3) f32, qff_vector (36, 32, 80, 1) f32, freqs (6) f32
//   out (N, 48) f32
// Strategy:
//   1) repack table to f16, layout (m=36, q=80, cr=32)  -> 184320 B in d_ws
//   2) main kernel: TDM tensor_load_to_lds stages the table into LDS with
//      hardware row padding (64B data + 16B pad -> 80B stride) to spread the
//      per-lane gather rows across 16 bank offsets instead of 4.
//   3) per-lane 2-tap gather (ds_load_b128), packed v_pk f16 lerp/CP-product,
//      non-temporal float4 stores for the 25MB output stream.
// ---------------------------------------------------------------------------

#define NFREQ 6
#define NM    36                    // NFREQ * 2 * 3
#define NQ    80
#define NCR   32                    // C(4) * R(8)
#define NROWS (NM * NQ)             // 2880 rows of 32 f16 (64B)
#define TBL_ELEMS (NROWS * NCR)     // 92160 halves in global (unpadded)
#define ROW_BYTES 80                // 64B data + 16B TDM pad in LDS
#define LDS_BYTES (NROWS * ROW_BYTES)   // 230400 B  (< 320KB WGP LDS)

typedef unsigned int       u32;
typedef unsigned long long u64;
typedef u32   u32x4 __attribute__((ext_vector_type(4)));
typedef int   i32x4 __attribute__((ext_vector_type(4)));
typedef int   i32x8 __attribute__((ext_vector_type(8)));
typedef float f32x4 __attribute__((ext_vector_type(4)));

// ---- kernel 1: (m, cr, q) f32  ->  (m, q, cr) f16 ------------------------
// grid (36, 10) x 256: blockIdx.x = m, 2560 (q,cr) pairs per m.
__global__ void qff_repack(const float* __restrict__ qv,
                           __half* __restrict__ out) {
    int m   = blockIdx.x;
    int idx = blockIdx.y * blockDim.x + threadIdx.x;   // 0..2559
    int q   = idx >> 5;
    int cr  = idx & (NCR - 1);
    out[(m * NQ + q) * NCR + cr] = __float2half(qv[(m * NCR + cr) * NQ + q]);
}

// ---- kernel 2: main ------------------------------------------------------
__global__ __launch_bounds__(256) void qff_main(const float* __restrict__ pts,
                                                const __half* __restrict__ tbl_g,
                                                const float* __restrict__ freqs,
                                                float* __restrict__ out,
                                                int npts) {
    extern __shared__ __half tbl[];

    // -------- stage the table into LDS (row-padded) ----------------------
#if defined(__gfx1250__) && __has_builtin(__builtin_amdgcn_tensor_load_to_lds)
    if (threadIdx.x == 0) {
        u64 ga = (u64)(const void*)tbl_g;
        u32 lds_off = (u32)(uintptr_t)(void*)tbl;   // low 32 bits = LDS offset
        // D# group0: count=1 | lds_addr | global_addr(57b) | type=2
        u32x4 g0 = { 1u,
                     lds_off,
                     (u32)ga,
                     (u32)((ga >> 32) & 0x1FFFFFFu) | (2u << 30) };
        // D# group1: 2D tensor, data_size=2B, tile 32 x 2880 (whole table),
        // pad_enable, pad_interval=3 (every 16 DWORDs = one 64B row),
        // pad_amount=3 (4 DWORDs = 16B) -> LDS row stride 80B.
        i32x8 g1 = { (int)((1u << 16) | (1u << 20) | (3u << 22) | (3u << 25)),
                     (int)((u32)NCR << 16),        // tensor_dim0 lo16 = 32
                     (int)((u32)NROWS << 16),      // tensor_dim1 lo16 = 2880
                     (int)((u32)NCR << 16),        // tile_dim0 = 32 (hi16)
                     (int)NROWS,                   // tile_dim1 = 2880 (lo16)
                     (int)NCR,                     // tensor_dim0_stride = 32
                     0, 0 };
        i32x4 gz4 = { 0, 0, 0, 0 };
        i32x8 gz8 = { 0, 0, 0, 0, 0, 0, 0, 0 };
        __builtin_amdgcn_tensor_load_to_lds(g0, g1, gz4, gz4, gz8, 0);
        __builtin_amdgcn_s_wait_tensorcnt(0);
    }
#else
    // cooperative fallback staging with manual 16B row padding
    {
        const uint4* src = (const uint4*)tbl_g;
        for (int i = threadIdx.x; i < NROWS * 4; i += blockDim.x) {
            int row = i >> 2, k = i & 3;
            *(uint4*)((char*)tbl + row * ROW_BYTES + k * 16) = src[i];
        }
    }
#endif
    __syncthreads();

    int pt = blockIdx.x * blockDim.x + threadIdx.x;
    if (pt < npts) {
        float p[3];
        p[0] = pts[3 * pt + 0];
        p[1] = pts[3 * pt + 1];
        p[2] = pts[3 * pt + 2];

        float acc[12][4];

        const float QS = 0.5f * (float)(NQ - 1);   // 39.5

        for (int fq = 0; fq < NFREQ; ++fq) {
            float fr = freqs[fq];
            float sn[3], cs[3];
#pragma unroll
            for (int d = 0; d < 3; ++d)
                __sincosf(p[d] * fr, &sn[d], &cs[d]);

#pragma unroll
            for (int s = 0; s < 2; ++s) {
                int fc = fq * 2 + s;
                u32 addr[3];
                __half2 w2[3];
#pragma unroll
                for (int d = 0; d < 3; ++d) {
                    float enc = s ? cs[d] : sn[d];
                    float pos = (enc + 1.0f) * QS;        // [0, 79]
                    float fi  = floorf(pos);
                    fi = fminf(fmaxf(fi, 0.0f), (float)(NQ - 2));
                    float w = pos - fi;                   // lerp weight
                    u32 row = (u32)(fc * 3 + d) * NQ + (u32)(int)fi;
                    addr[d] = row * ROW_BYTES;
                    w2[d] = __float2half2_rn(w);
                }
#pragma unroll
                for (int c = 0; c < 4; ++c) {             // 4 feats, 8 ranks each
                    __half2 prod[4];
#pragma unroll
                    for (int d = 0; d < 3; ++d) {
                        const char* b = (const char*)tbl + addr[d] + c * 16;
                        uint4 u0 = *(const uint4*)(b);              // q = i0
                        uint4 u1 = *(const uint4*)(b + ROW_BYTES);  // q = i0+1
#pragma unroll
                        for (int k = 0; k < 4; ++k) {
                            __half2 v0 = __builtin_bit_cast(__half2, (&u0.x)[k]);
                            __half2 v1 = __builtin_bit_cast(__half2, (&u1.x)[k]);
                            // v0 + w*(v1-v0)  -> v_pk_fma_f16
                            __half2 v = __hfma2(w2[d], __hsub2(v1, v0), v0);
                            prod[k] = (d == 0) ? v : __hmul2(prod[k], v);
                        }
                    }
                    __half2 s2 = __hadd2(__hadd2(prod[0], prod[1]),
                                         __hadd2(prod[2], prod[3]));
                    acc[fc][c] = __low2float(s2) + __high2float(s2);
                }
            }
        }

        // out[pt][fc*4+c]: 12 non-temporal float4 stores (25MB output stream)
        f32x4* o = (f32x4*)(out + (size_t)pt * 48);
#pragma unroll
        for (int fc = 0; fc < 12; ++fc) {
            f32x4 v = { acc[fc][0], acc[fc][1], acc[fc][2], acc[fc][3] };
            __builtin_nontemporal_store(v, &o[fc]);
        }
    }
}

// ---------------------------------------------------------------------------
extern "C" void kernel_launch(void* const* d_in, const int* in_sizes, int n_in,
                              void* d_out, int out_size, void* d_ws, size_t ws_size,
                              hipStream_t stream) {
    const float* points = (const float*)d_in[0];
    const float* qff    = (const float*)d_in[1];
    const float* freqs  = (const float*)d_in[2];
    float*  out  = (float*)d_out;
    __half* tblw = (__half*)d_ws;
    int npts = in_sizes[0] / 3;

    // allow >64KB dynamic LDS (CDNA5 WGP has 320KB)
    (void)hipFuncSetAttribute((const void*)qff_main,
                              hipFuncAttributeMaxDynamicSharedMemorySize,
                              LDS_BYTES);

    qff_repack<<<dim3(NM, (NQ * NCR) / 256), 256, 0, stream>>>(qff, tblw);
    qff_main<<<(npts + 255) / 256, 256, LDS_BYTES, stream>>>(points, tblw, freqs,
                                                             out, npts);
}